// GraphAttentionLayer_54107997995611
// MI455X (gfx1250) — compile-verified
//
#include <hip/hip_runtime.h>
#include <hip/hip_bf16.h>

#define IN_F 128
#define OUT_F 64
#define NEG_SLOPE 0.2f
#define LDS_STRIDE 130   // 128 + 2 pad: row stride % 64 banks == 2 -> conflict-free column reads

typedef float v2f __attribute__((ext_vector_type(2)));
typedef float v8f __attribute__((ext_vector_type(8)));

// ---- sortable-uint encoding of float for integer atomicMax ----
__device__ __forceinline__ unsigned fenc(float f) {
    unsigned b = __float_as_uint(f);
    return (b & 0x80000000u) ? ~b : (b | 0x80000000u);
}
__device__ __forceinline__ float fdec(unsigned u) {
    return __uint_as_float((u & 0x80000000u) ? (u & 0x7FFFFFFFu) : ~u);
}

// ============================================================
// Kernel 1: h_lin = h @ W^T   via V_WMMA_F32_16X16X4_F32
// block = 256 threads = 8 waves; block computes 32 rows x 64 cols
// wave w: row tile = w/4 (16 rows), col tile = w%4 (16 cols)
// ============================================================
__global__ __launch_bounds__(256)
void gat_gemm_wmma(const float* __restrict__ h, const float* __restrict__ W,
                   float* __restrict__ h_lin, int n_nodes) {
    __shared__ float sh[32 * LDS_STRIDE];   // h tile  32x128 (padded)
    __shared__ float sw[64 * LDS_STRIDE];   // W full  64x128 (padded)

    const int tid  = threadIdx.x;
    const int wave = tid >> 5;
    const int lane = tid & 31;
    const int row0 = blockIdx.x * 32;

    // stage W (64x128) into LDS, coalesced
    for (int i = tid; i < 64 * IN_F; i += 256) {
        int r = i >> 7, c = i & 127;
        sw[r * LDS_STRIDE + c] = W[i];
    }
    // stage h tile (32x128) into LDS, coalesced
    for (int i = tid; i < 32 * IN_F; i += 256) {
        int r = i >> 7, c = i & 127;
        int gr = row0 + r;
        sh[r * LDS_STRIDE + c] = (gr < n_nodes) ? h[(size_t)gr * IN_F + c] : 0.0f;
    }
    __syncthreads();

    const int rt   = wave >> 2;   // 0..1
    const int ct   = wave & 3;    // 0..3
    const int half = lane >> 4;   // K sub-block select (lanes 16-31 hold K=2,3)
    const int l16  = lane & 15;

    // A fragment: lane l<16 -> h_tile[M=l][k0+0..1]; lane>=16 -> [M=l-16][k0+2..3]
    const float* pa = &sh[(rt * 16 + l16) * LDS_STRIDE + half * 2];
    // B fragment (B = W^T): lane l<16 -> W[N=l][k0+0..1]; lane>=16 -> W[N=l-16][k0+2..3]
    const float* pb = &sw[(ct * 16 + l16) * LDS_STRIDE + half * 2];

    v8f c = {};
#pragma unroll
    for (int k = 0; k < IN_F; k += 4) {
        v2f a = *(const v2f*)(pa + k);
        v2f b = *(const v2f*)(pb + k);
        c = __builtin_amdgcn_wmma_f32_16x16x4_f32(
                /*neg_a=*/false, a, /*neg_b=*/false, b,
                /*c_mod=*/(short)0, c, /*reuse_a=*/false, /*reuse_b=*/false);
    }

    // C/D layout: VGPR r: lanes 0-15 = row r, lanes 16-31 = row r+8; col = lane&15
    const int rbase = row0 + rt * 16;
#pragma unroll
    for (int r = 0; r < 8; ++r) {
        int rr = rbase + (half ? (r + 8) : r);
        if (rr < n_nodes)
            h_lin[(size_t)rr * OUT_F + ct * 16 + l16] = c[r];
    }
}

// ============================================================
// Kernel 2: per-node scores  s_src = h_lin . a1, s_dst = h_lin . a2
// one wave32 per node
// ============================================================
__global__ __launch_bounds__(256)
void gat_scores(const float* __restrict__ h_lin, const float* __restrict__ a,
                float* __restrict__ s_src, float* __restrict__ s_dst, int n) {
    int node = (blockIdx.x * blockDim.x + threadIdx.x) >> 5;
    int lane = threadIdx.x & 31;
    if (node >= n) return;
    const float* hv = h_lin + (size_t)node * OUT_F;
    float v0 = hv[lane], v1 = hv[lane + 32];
    float p1 = v0 * a[lane]      + v1 * a[lane + 32];   // a1 = a[0:64]
    float p2 = v0 * a[64 + lane] + v1 * a[96 + lane];   // a2 = a[64:128]
#pragma unroll
    for (int off = 16; off > 0; off >>= 1) {
        p1 += __shfl_xor(p1, off, 32);
        p2 += __shfl_xor(p2, off, 32);
    }
    if (lane == 0) { s_src[node] = p1; s_dst[node] = p2; }
}

// ============================================================
// Kernel 3: init output accumulator, denom, and segment-max
// ============================================================
__global__ void gat_init(float* __restrict__ out, float* __restrict__ denom,
                         unsigned* __restrict__ m, int n) {
    int i = blockIdx.x * blockDim.x + threadIdx.x;
    if (i < n * OUT_F) out[i] = 0.0f;
    if (i < n) { denom[i] = 0.0f; m[i] = fenc(-__builtin_inff()); }
}

__device__ __forceinline__ void edge_rc(const int* __restrict__ ei, int i, int E,
                                        int& row, int& col) {
    if (i < E) { row = ei[i]; col = ei[E + i]; }
    else       { row = col = i - E; }   // self loops appended
}

// ============================================================
// Kernel 4: segment max of leakyrelu(s_src[row]+s_dst[col]) over row
// ============================================================
__global__ __launch_bounds__(256)
void gat_edge_max(const int* __restrict__ ei, const float* __restrict__ s_src,
                  const float* __restrict__ s_dst, unsigned* __restrict__ m,
                  int E, int total) {
    int i = blockIdx.x * blockDim.x + threadIdx.x;
    if (i >= total) return;
    int row, col; edge_rc(ei, i, E, row, col);
    float e = s_src[row] + s_dst[col];
    e = e > 0.0f ? e : NEG_SLOPE * e;
    atomicMax(&m[row], fenc(e));   // native GLOBAL_ATOMIC_MAX_U32
}

// ============================================================
// Kernel 5: ex = exp(e - m[row]); denom[row] += ex; stash ex per edge
// ============================================================
__global__ __launch_bounds__(256)
void gat_edge_exp(const int* __restrict__ ei, const float* __restrict__ s_src,
                  const float* __restrict__ s_dst, const unsigned* __restrict__ m,
                  float* __restrict__ denom, float* __restrict__ exv,
                  int E, int total) {
    int i = blockIdx.x * blockDim.x + threadIdx.x;
    if (i >= total) return;
    int row, col; edge_rc(ei, i, E, row, col);
    float e = s_src[row] + s_dst[col];
    e = e > 0.0f ? e : NEG_SLOPE * e;
    float ex = __expf(e - fdec(m[row]));
    exv[i] = ex;
    unsafeAtomicAdd(&denom[row], ex);  // native GLOBAL_ATOMIC_ADD_F32
}

// ============================================================
// Kernel 6: out[row] += (ex/denom[row]) * h_lin[col]
// 64 threads per edge (one per feature); 4 edges per 256-thread block
// ============================================================
__global__ __launch_bounds__(256)
void gat_scatter(const int* __restrict__ ei, const float* __restrict__ h_lin,
                 const float* __restrict__ exv, const float* __restrict__ denom,
                 float* __restrict__ out, int E, int total) {
    int t    = blockIdx.x * blockDim.x + threadIdx.x;
    int edge = t >> 6;
    int f    = t & 63;
    if (edge >= total) return;
    int row, col; edge_rc(ei, edge, E, row, col);
    float alpha = exv[edge] / denom[row];
    float v = h_lin[(size_t)col * OUT_F + f];   // L2-resident gather
    unsafeAtomicAdd(&out[(size_t)row * OUT_F + f], alpha * v);
}

// ============================================================
// Kernel 7: in-place ELU
// ============================================================
__global__ void gat_elu(float* __restrict__ out, int total) {
    int i = blockIdx.x * blockDim.x + threadIdx.x;
    if (i >= total) return;
    float x = out[i];
    out[i] = x > 0.0f ? x : (__expf(x) - 1.0f);
}

extern "C" void kernel_launch(void* const* d_in, const int* in_sizes, int n_in,
                              void* d_out, int out_size, void* d_ws, size_t ws_size,
                              hipStream_t stream) {
    const float* h  = (const float*)d_in[0];   // (N,128)
    const float* W  = (const float*)d_in[1];   // (64,128)
    const float* a  = (const float*)d_in[2];   // (1,128)
    const int*   ei = (const int*)d_in[3];     // (2,E) row-major

    const int N = in_sizes[0] / IN_F;
    const int E = in_sizes[3] / 2;
    const int total = E + N;                   // edges + self loops

    // workspace layout
    float*    h_lin = (float*)d_ws;                          // N*64
    float*    s_src = h_lin + (size_t)N * OUT_F;             // N
    float*    s_dst = s_src + N;                             // N
    float*    denom = s_dst + N;                             // N
    unsigned* m     = (unsigned*)(denom + N);                // N
    float*    exv   = (float*)(m + N);                       // E+N

    float* out = (float*)d_out;

    gat_gemm_wmma<<<(N + 31) / 32, 256, 0, stream>>>(h, W, h_lin, N);
    gat_scores  <<<(N * 32 + 255) / 256, 256, 0, stream>>>(h_lin, a, s_src, s_dst, N);
    gat_init    <<<(N * OUT_F + 255) / 256, 256, 0, stream>>>(out, denom, m, N);
    gat_edge_max<<<(total + 255) / 256, 256, 0, stream>>>(ei, s_src, s_dst, m, E, total);
    gat_edge_exp<<<(total + 255) / 256, 256, 0, stream>>>(ei, s_src, s_dst, m, denom, exv, E, total);
    gat_scatter <<<((size_t)total * 64 + 255) / 256, 256, 0, stream>>>(ei, h_lin, exv, denom, out, E, total);
    gat_elu     <<<(N * OUT_F + 255) / 256, 256, 0, stream>>>(out, N * OUT_F);
}